// Mp_encoder_53592601919692
// MI455X (gfx1250) — compile-verified
//
#include <hip/hip_runtime.h>
#include <math.h>

// ---------------------------------------------------------------------------
// 3-layer GraphConv (DGL norm='both' + edge weight) for MI455X / gfx1250.
//  N=10000 nodes, E=160000 edges, D=512.
//  GEMM uses V_WMMA_F32_16X16X4_F32 (full fp32 precision, wave32 WMMA).
// ---------------------------------------------------------------------------

#define DH 512
#define WPB 8            // waves per block (256 threads, wave32)

typedef float v2f __attribute__((ext_vector_type(2)));
typedef float v8f __attribute__((ext_vector_type(8)));

// ---------------- deterministic stateless dropout hash ---------------------
__device__ __forceinline__ float urand01(unsigned h) {
  h ^= h >> 16; h *= 0x7FEB352Du;
  h ^= h >> 15; h *= 0x846CA68Bu;
  h ^= h >> 16;
  return (float)(h >> 8) * (1.0f / 16777216.0f);
}
__device__ __forceinline__ float dropf(float x, unsigned seed, unsigned idx) {
  unsigned h = idx * 0x9E3779B9u ^ (seed * 0x85EBCA6Bu + 0xC2B2AE35u);
  // keep prob = 0.8, scale kept values by 1/0.8
  return (urand01(h) < 0.8f) ? x * 1.25f : 0.0f;
}

// ---------------- small utility kernels ------------------------------------
__global__ void fill_zero(float* __restrict__ p, int n) {
  int i = blockIdx.x * blockDim.x + threadIdx.x;
  if (i < n) p[i] = 0.0f;
}

__global__ void degree_kernel(const int* __restrict__ src, const int* __restrict__ dst,
                              float* __restrict__ deg_out, float* __restrict__ deg_in, int E) {
  int e = blockIdx.x * blockDim.x + threadIdx.x;
  if (e < E) {
    atomicAdd(&deg_out[src[e]], 1.0f);
    atomicAdd(&deg_in[dst[e]], 1.0f);
  }
}

__global__ void rsqrt_kernel(float* __restrict__ p, int n) {
  int i = blockIdx.x * blockDim.x + threadIdx.x;
  if (i < n) p[i] = rsqrtf(fmaxf(p[i], 1.0f));
}

// ------------- fused dropout + src-norm + gather + scatter-add -------------
// one thread per (edge, 4-feature group): E * 128 threads
__global__ void scatter_kernel(const float* __restrict__ X, const int* __restrict__ src,
                               const int* __restrict__ dst, const float* __restrict__ ew,
                               const float* __restrict__ dnorm_out, float* __restrict__ agg,
                               int E, unsigned seed) {
  int tid = blockIdx.x * blockDim.x + threadIdx.x;
  if (tid >= E * (DH / 4)) return;
  int e = tid >> 7;               // DH/4 == 128 groups per edge
  int d = (tid & 127) << 2;
  int s = src[e];
  float w = ew[e] * dnorm_out[s];
  const float4 x = *(const float4*)(X + (size_t)s * DH + d);
  unsigned base = (unsigned)s * DH + (unsigned)d;
  float4 v;
  v.x = dropf(x.x, seed, base + 0) * w;
  v.y = dropf(x.y, seed, base + 1) * w;
  v.z = dropf(x.z, seed, base + 2) * w;
  v.w = dropf(x.w, seed, base + 3) * w;
  float* out = agg + (size_t)dst[e] * DH + d;
  atomicAdd(out + 0, v.x);
  atomicAdd(out + 1, v.y);
  atomicAdd(out + 2, v.z);
  atomicAdd(out + 3, v.w);
}

// ------------- WMMA fp32 GEMM + dst-norm + bias + optional ELU -------------
// One wave computes a 16(M) x 64(N) strip: A-fragment reused across 4 WMMAs.
// A: agg [M x 512] row-major, W: [512 x 512] row-major.
__global__ void __launch_bounds__(256)
gemm_wmma(const float* __restrict__ A, const float* __restrict__ W,
          const float* __restrict__ bias, const float* __restrict__ dnorm_in,
          float* __restrict__ Y, int M, int do_elu) {
  const int lane   = threadIdx.x & 31;
  const int wave   = threadIdx.x >> 5;
  const int waveId = blockIdx.x * WPB + wave;
  const int NSTRIPS = DH / 64;              // 8 strips of 64 columns
  const int mtile  = waveId / NSTRIPS;
  const int nstrip = waveId % NSTRIPS;
  if (mtile * 16 >= M) return;              // wave-uniform: EXEC stays all-1s
  const int m0   = mtile * 16;
  const int n0   = nstrip * 64;
  const int half = lane >> 4;               // 0: K={0,1}, 1: K={2,3}
  const int l16  = lane & 15;

  v8f acc0 = {}, acc1 = {}, acc2 = {}, acc3 = {};

  const float* arow = A + (size_t)(m0 + l16) * DH + 2 * half;
  const float* wb   = W + n0 + l16;

  for (int k = 0; k < DH; k += 4) {
    // A fragment (16x4 f32): lane l16 = row, VGPR0/1 = K pair per lane half
    const float2 av = *(const float2*)(arow + k);
    v2f a; a.x = av.x; a.y = av.y;

    const int kr = (k + 2 * half) * DH;     // row (k + 2*half) of W
    v2f b0, b1, b2, b3;
    b0.x = wb[kr +  0]; b0.y = wb[kr + DH +  0];
    b1.x = wb[kr + 16]; b1.y = wb[kr + DH + 16];
    b2.x = wb[kr + 32]; b2.y = wb[kr + DH + 32];
    b3.x = wb[kr + 48]; b3.y = wb[kr + DH + 48];

    acc0 = __builtin_amdgcn_wmma_f32_16x16x4_f32(false, a, false, b0, (short)0, acc0, false, false);
    acc1 = __builtin_amdgcn_wmma_f32_16x16x4_f32(false, a, false, b1, (short)0, acc1, false, false);
    acc2 = __builtin_amdgcn_wmma_f32_16x16x4_f32(false, a, false, b2, (short)0, acc2, false, false);
    acc3 = __builtin_amdgcn_wmma_f32_16x16x4_f32(false, a, false, b3, (short)0, acc3, false, false);
  }

  // C/D layout: VGPR r -> M = r (lanes 0-15) / r+8 (lanes 16-31), N = l16
  #pragma unroll
  for (int r = 0; r < 8; ++r) {
    const int row = m0 + r + 8 * half;
    const float dn = dnorm_in[row];
    float* yrow = Y + (size_t)row * DH;
    const int c0 = n0 + l16;
    float v0 = acc0[r] * dn + bias[c0 +  0];
    float v1 = acc1[r] * dn + bias[c0 + 16];
    float v2 = acc2[r] * dn + bias[c0 + 32];
    float v3 = acc3[r] * dn + bias[c0 + 48];
    if (do_elu) {
      v0 = v0 > 0.0f ? v0 : expm1f(v0);
      v1 = v1 > 0.0f ? v1 : expm1f(v1);
      v2 = v2 > 0.0f ? v2 : expm1f(v2);
      v3 = v3 > 0.0f ? v3 : expm1f(v3);
    }
    yrow[c0 +  0] = v0;
    yrow[c0 + 16] = v1;
    yrow[c0 + 32] = v2;
    yrow[c0 + 48] = v3;
  }
}

// ---------------------------------------------------------------------------
extern "C" void kernel_launch(void* const* d_in, const int* in_sizes, int n_in,
                              void* d_out, int out_size, void* d_ws, size_t ws_size,
                              hipStream_t stream) {
  const float* h  = (const float*)d_in[0];
  const int*   src = (const int*)d_in[1];
  const int*   dst = (const int*)d_in[2];
  const float* ew  = (const float*)d_in[3];
  const float* W0  = (const float*)d_in[4];
  const float* b0  = (const float*)d_in[5];
  const float* W1  = (const float*)d_in[6];
  const float* b1  = (const float*)d_in[7];
  const float* W2  = (const float*)d_in[8];
  const float* b2  = (const float*)d_in[9];

  const int N = in_sizes[0] / DH;   // 10000
  const int E = in_sizes[1];        // 160000
  const int ND = N * DH;

  float* ws        = (float*)d_ws;
  float* dnorm_out = ws;                 // [N]
  float* dnorm_in  = ws + N;             // [N]
  float* agg       = ws + 2 * N;         // [N*DH]
  float* y         = agg + (size_t)ND;   // [N*DH]
  float* out       = (float*)d_out;

  const int T = 256;

  // degrees -> rsqrt norms (computed fresh every call: deterministic)
  fill_zero<<<(2 * N + T - 1) / T, T, 0, stream>>>(dnorm_out, 2 * N);
  degree_kernel<<<(E + T - 1) / T, T, 0, stream>>>(src, dst, dnorm_out, dnorm_in, E);
  rsqrt_kernel<<<(2 * N + T - 1) / T, T, 0, stream>>>(dnorm_out, 2 * N);

  const int scat_threads = E * (DH / 4);
  const int scat_blocks  = (scat_threads + T - 1) / T;
  const int gemm_blocks  = ((N + 15) / 16 * (DH / 64) + WPB - 1) / WPB;

  // layer 0: h -> y  (ELU)
  fill_zero<<<(ND + T - 1) / T, T, 0, stream>>>(agg, ND);
  scatter_kernel<<<scat_blocks, T, 0, stream>>>(h, src, dst, ew, dnorm_out, agg, E, 1u);
  gemm_wmma<<<gemm_blocks, T, 0, stream>>>(agg, W0, b0, dnorm_in, y, N, 1);

  // layer 1: y -> y  (ELU)
  fill_zero<<<(ND + T - 1) / T, T, 0, stream>>>(agg, ND);
  scatter_kernel<<<scat_blocks, T, 0, stream>>>(y, src, dst, ew, dnorm_out, agg, E, 2u);
  gemm_wmma<<<gemm_blocks, T, 0, stream>>>(agg, W1, b1, dnorm_in, y, N, 1);

  // layer 2: y -> out (no activation)
  fill_zero<<<(ND + T - 1) / T, T, 0, stream>>>(agg, ND);
  scatter_kernel<<<scat_blocks, T, 0, stream>>>(y, src, dst, ew, dnorm_out, agg, E, 3u);
  gemm_wmma<<<gemm_blocks, T, 0, stream>>>(agg, W2, b2, dnorm_in, out, N, 0);
}